// GaussianHMM_37117107372254
// MI455X (gfx1250) — compile-verified
//
#include <hip/hip_runtime.h>

#define NS 32      // states
#define ND 16      // dims
#define NB 32      // batch
#define TM 2048    // T_MAX

typedef __attribute__((ext_vector_type(2))) float v2f;
typedef __attribute__((ext_vector_type(4))) float v4f;
typedef __attribute__((ext_vector_type(8))) float v8f;

// Device-global scratch (fully rewritten every call -> deterministic, no ws_size dependence)
__device__ float g_A[NS * NS];        // exp(log_softmax(trans, axis=1)), row-major [j][k]
__device__ float g_logpi[NS];
__device__ float g_Pt[NS * ND * ND];  // per state: Pt[k][i] = P[i][k], P = L^-1 (lower tri)
__device__ float g_const[NS];         // -0.5*(D*log(2pi) + logdet(Sigma))

// ---------------------------------------------------------------------------
// Setup: softmaxes + per-state triangular inversion (one-time, tiny)
// ---------------------------------------------------------------------------
__global__ __launch_bounds__(128) void hmm_setup(const float* __restrict__ pi,
                                                 const float* __restrict__ trans,
                                                 const float* __restrict__ chol) {
    const int tid = threadIdx.x;
    if (tid < NS) {
        // Invert L_n (forward substitution), n = tid
        const int n = tid;
        float L[ND][ND];
        float diagsum = 0.0f;
        for (int i = 0; i < ND; ++i) {
            for (int j = 0; j < ND; ++j) {
                float c = chol[(n * ND + i) * ND + j];
                if (i == j)      { diagsum += c; L[i][j] = __expf(c); }
                else if (i > j)  { L[i][j] = c; }
                else             { L[i][j] = 0.0f; }
            }
        }
        // log(2*pi) = 1.8378770664093453
        g_const[n] = -0.5f * (16.0f * 1.8378770664093453f + 2.0f * diagsum);
        for (int c = 0; c < ND; ++c) {       // solve L y = e_c  -> column c of P
            float y[ND];
            for (int i = 0; i < ND; ++i) {
                float s = (i == c) ? 1.0f : 0.0f;
                for (int j = c; j < i; ++j) s -= L[i][j] * y[j];
                y[i] = (i < c) ? 0.0f : (s / L[i][i]);
            }
            for (int i = 0; i < ND; ++i)
                g_Pt[n * (ND * ND) + c * ND + i] = y[i];   // Pt[k=c][i] = P[i][c]
        }
    } else if (tid < 2 * NS) {
        // Row j of transition matrix: linear softmax (axis=1)
        const int j = tid - NS;
        float m = -INFINITY;
        for (int k = 0; k < NS; ++k) m = fmaxf(m, trans[j * NS + k]);
        float s = 0.0f;
        for (int k = 0; k < NS; ++k) s += __expf(trans[j * NS + k] - m);
        const float lse = m + __logf(s);
        for (int k = 0; k < NS; ++k) g_A[j * NS + k] = __expf(trans[j * NS + k] - lse);
    } else if (tid == 2 * NS) {
        float m = -INFINITY;
        for (int k = 0; k < NS; ++k) m = fmaxf(m, pi[k]);
        float s = 0.0f;
        for (int k = 0; k < NS; ++k) s += __expf(pi[k] - m);
        const float lse = m + __logf(s);
        for (int k = 0; k < NS; ++k) g_logpi[k] = pi[k] - lse;
    }
}

// ---------------------------------------------------------------------------
// Emission: logB[b,t,n] = const_n - 0.5*||P_n (x - mu_n)||^2, written to d_out.
// One wave per (state, group of tiles). Y = Delta(16x16) * Pt(16x16) via
// four chained V_WMMA_F32_16X16X4_F32 (K = 16), then a staged 16-lane
// butterfly (8 independent bpermutes per stage) and one predicated store.
// ---------------------------------------------------------------------------
#define TILES_PER_WAVE 16

__global__ __launch_bounds__(32) void hmm_emission(const float* __restrict__ x,
                                                   const float* __restrict__ means,
                                                   float* __restrict__ out) {
    const int l  = threadIdx.x;      // 0..31
    const int h  = l >> 4;           // half of wave
    const int ln = l & 15;
    const int n  = blockIdx.y;       // state

    // B operand: Pt chunks. B[k][i]; lane: N=i=ln, K = 4c + 2h + {0,1}
    v2f Bop[4], muv[4];
#pragma unroll
    for (int c = 0; c < 4; ++c) {
        const int k0 = 4 * c + 2 * h;
        Bop[c].x = g_Pt[n * 256 + (k0 + 0) * ND + ln];
        Bop[c].y = g_Pt[n * 256 + (k0 + 1) * ND + ln];
        muv[c]   = *(const v2f*)(means + n * ND + k0);
    }
    const float ck = g_const[n];

    const int bt_base = blockIdx.x * (TILES_PER_WAVE * 16);
    for (int it = 0; it < TILES_PER_WAVE; ++it) {
        const int row0 = bt_base + it * 16;
        const int row  = row0 + ln;          // A operand: M = ln
        v2f Aop[4];
#pragma unroll
        for (int c = 0; c < 4; ++c) {
            const int k0 = 4 * c + 2 * h;
            v2f xv = *(const v2f*)(x + row * ND + k0);
            Aop[c] = xv - muv[c];
        }
        v8f acc = {};
#pragma unroll
        for (int c = 0; c < 4; ++c)
            acc = __builtin_amdgcn_wmma_f32_16x16x4_f32(false, Aop[c], false, Bop[c],
                                                        (short)0, acc, false, false);
        // quad per tile row: staged butterfly over the 16 N-lanes of each half.
        // Stage-major order -> 8 independent bpermutes in flight per stage.
        float q[8];
#pragma unroll
        for (int r = 0; r < 8; ++r) q[r] = acc[r] * acc[r];
#pragma unroll
        for (int stage = 1; stage <= 8; stage <<= 1) {
#pragma unroll
            for (int r = 0; r < 8; ++r) q[r] += __shfl_xor(q[r], stage, 32);
        }
        // Lane (b2 b1 b0) selects q[ln&7]; lanes with (ln&8)==0 store rows
        // row0 + h*8 + (ln&7)  (half0 butterflies hold rows r, half1 rows 8+r).
        float v01 = (ln & 1) ? q[1] : q[0];
        float v23 = (ln & 1) ? q[3] : q[2];
        float v45 = (ln & 1) ? q[5] : q[4];
        float v67 = (ln & 1) ? q[7] : q[6];
        float v03 = (ln & 2) ? v23 : v01;
        float v47 = (ln & 2) ? v67 : v45;
        float v   = (ln & 4) ? v47 : v03;
        if ((ln & 8) == 0)
            out[(size_t)(row0 + h * 8 + (ln & 7)) * NS + n] = ck - 0.5f * v;
    }
}

// ---------------------------------------------------------------------------
// Forward recursion, in place over d_out (which holds logB on entry).
// Grid = 2 blocks x 1 wave; block bh owns batches [bh*16, bh*16+16).
// Per step: S[j,b] = sum_k A[j,k] * exp(alpha[b,k]-m_b)  via 16 WMMAs,
// then alpha[b,j] = log(S) + m_b + logB[b,t,j]. Single wave -> no barriers.
// m_b kept as a running per-lane register max (lane + xor16 partner own the
// full alpha row), so only the half-exchange goes through LDS.
// ---------------------------------------------------------------------------
__global__ __launch_bounds__(32) void hmm_forward(float* __restrict__ out) {
    __shared__ __align__(16) float alpha[16 * NS];   // [b_local][j]
    const int l  = threadIdx.x;
    const int h  = l >> 4;
    const int ln = l & 15;                 // b_local (N-index) for B/C/D operands
    const int b  = blockIdx.x * 16 + ln;

    // Persistent A operand: Aop[jh][c]; lane: M = ln -> j = jh*16+ln, K = 4c+2h+{0,1}
    v2f Aop[2][8];
#pragma unroll
    for (int jh = 0; jh < 2; ++jh)
#pragma unroll
        for (int c = 0; c < 8; ++c) {
            const int j  = jh * 16 + ln;
            const int k0 = 4 * c + 2 * h;
            Aop[jh][c] = *(const v2f*)(&g_A[j * NS + k0]);
        }

    float* outb = out + (size_t)b * TM * NS;

    // t = 0: alpha0 = logB + log_pi (in place), seed LDS, seed running max
    float lmax = -INFINITY;
#pragma unroll
    for (int jh = 0; jh < 2; ++jh) {
        const int j0 = jh * 16 + h * 8;
        v4f lo = *(const v4f*)(outb + j0)     + *(const v4f*)(g_logpi + j0);
        v4f hi = *(const v4f*)(outb + j0 + 4) + *(const v4f*)(g_logpi + j0 + 4);
        *(v4f*)(outb + j0)                = lo;
        *(v4f*)(outb + j0 + 4)            = hi;
        *(v4f*)(&alpha[ln * NS + j0])     = lo;
        *(v4f*)(&alpha[ln * NS + j0 + 4]) = hi;
        lmax = fmaxf(lmax, fmaxf(fmaxf(lo.x, lo.y), fmaxf(lo.z, lo.w)));
        lmax = fmaxf(lmax, fmaxf(fmaxf(hi.x, hi.y), fmaxf(hi.z, hi.w)));
    }

    for (int t = 1; t < TM; ++t) {
        // m_b = max over this lane's 16 alpha entries and its xor16 partner's 16
        const float m = fmaxf(lmax, __shfl_xor(lmax, 16, 32));
        float* outrow = outb + (size_t)t * NS;
        if (t + 1 < TM) __builtin_prefetch(outrow + NS, 0, 3);   // next logB row (1 line)
        // B operand: E[k][b] = exp(alpha[b][k] - m); lane: N = ln, K = 4c+2h+{0,1}
        v2f Bop[8];
#pragma unroll
        for (int c = 0; c < 8; ++c) {
            const int k0 = 4 * c + 2 * h;
            v2f av = *(const v2f*)(&alpha[ln * NS + k0]);   // ds_load_b64
            Bop[c].x = __expf(av.x - m);
            Bop[c].y = __expf(av.y - m);
        }
        lmax = -INFINITY;
#pragma unroll
        for (int jh = 0; jh < 2; ++jh) {
            v8f acc = {};
#pragma unroll
            for (int c = 0; c < 8; ++c)
                acc = __builtin_amdgcn_wmma_f32_16x16x4_f32(false, Aop[jh][c], false, Bop[c],
                                                            (short)0, acc, false, false);
            const int j0 = jh * 16 + h * 8;                 // C vgpr r -> j = j0 + r
            v4f lbl = *(const v4f*)(outrow + j0);
            v4f lbh = *(const v4f*)(outrow + j0 + 4);
            v4f rl, rh;
            rl.x = lbl.x + m + __logf(acc[0]);
            rl.y = lbl.y + m + __logf(acc[1]);
            rl.z = lbl.z + m + __logf(acc[2]);
            rl.w = lbl.w + m + __logf(acc[3]);
            rh.x = lbh.x + m + __logf(acc[4]);
            rh.y = lbh.y + m + __logf(acc[5]);
            rh.z = lbh.z + m + __logf(acc[6]);
            rh.w = lbh.w + m + __logf(acc[7]);
            *(v4f*)(outrow + j0)              = rl;
            *(v4f*)(outrow + j0 + 4)          = rh;
            *(v4f*)(&alpha[ln * NS + j0])     = rl;
            *(v4f*)(&alpha[ln * NS + j0 + 4]) = rh;
            lmax = fmaxf(lmax, fmaxf(fmaxf(rl.x, rl.y), fmaxf(rl.z, rl.w)));
            lmax = fmaxf(lmax, fmaxf(fmaxf(rh.x, rh.y), fmaxf(rh.z, rh.w)));
        }
    }
}

// ---------------------------------------------------------------------------
extern "C" void kernel_launch(void* const* d_in, const int* in_sizes, int n_in,
                              void* d_out, int out_size, void* d_ws, size_t ws_size,
                              hipStream_t stream) {
    const float* x     = (const float*)d_in[0];  // (32, 2048, 16)
    const float* pi    = (const float*)d_in[1];  // (32,)
    const float* trans = (const float*)d_in[2];  // (32, 32)
    const float* means = (const float*)d_in[3];  // (32, 16)
    const float* chol  = (const float*)d_in[4];  // (32, 16, 16)
    // d_in[5] = T (int), unused by the reference computation
    float* out = (float*)d_out;                  // (32, 2048, 32) f32

    hmm_setup<<<dim3(1), dim3(128), 0, stream>>>(pi, trans, chol);

    const int n_tiles = (NB * TM) / 16;          // 4096 bt-tiles of 16 rows
    hmm_emission<<<dim3(n_tiles / TILES_PER_WAVE, NS), dim3(32), 0, stream>>>(x, means, out);

    hmm_forward<<<dim3(2), dim3(32), 0, stream>>>(out);
}